// HeliosSelfAttention_63290638074282
// MI455X (gfx1250) — compile-verified
//
#include <hip/hip_runtime.h>
#include <hip/hip_bf16.h>

typedef float v2f __attribute__((ext_vector_type(2)));
typedef float v8f __attribute__((ext_vector_type(8)));

#define NUM_HEADS 16
#define HEAD_DIM  64
#define DIM       1024
#define INNER     (NUM_HEADS * HEAD_DIM)   // 1024
#define QKV_N     (3 * INNER)              // 3072
#define BATCH     2
#define SEQ       2048
#define BS        (BATCH * SEQ)            // 4096
#define QTILES    (SEQ / 16)               // 128

// ---------------------------------------------------------------------------
// fp32 GEMM + bias using V_WMMA_F32_16X16X4_F32 with 64x64 register blocking.
// One wave computes a 64x64 C tile = 4x4 grid of 16x16 WMMA tiles
// (16 x v8f = 128 accumulator VGPRs). Per K-step of 4:
//   4 A-frags (global b64) + 4 B-frags (2x b32) -> 16 chained WMMAs,
// i.e. each loaded fragment is reused 4x by the matrix pipe.
// Layouts (ISA 7.12.2, 32-bit):
//   A-frag: lane l -> row m = l&15, K pair = 2*(l>>4)+{0,1}   (v2f)
//   B-frag: lane l -> col n = l&15, K pair = 2*(l>>4)+{0,1}   (v2f)
//   C/D   : VGPR r -> row r + 8*(l>>4), col l&15              (v8f)
// ---------------------------------------------------------------------------
__global__ __launch_bounds__(32)
void gemm_bias_f32(const float* __restrict__ A, const float* __restrict__ W,
                   const float* __restrict__ bias, float* __restrict__ C,
                   int M, int N, int K) {
    const int l    = threadIdx.x;
    const int lo   = l & 15;
    const int half = l >> 4;
    const int n0   = blockIdx.x * 64;
    const int m0   = blockIdx.y * 64;

    const float* arow[4];
    const float* wcol[4];
#pragma unroll
    for (int i = 0; i < 4; ++i)
        arow[i] = A + (size_t)(m0 + 16 * i + lo) * K + 2 * half;
#pragma unroll
    for (int j = 0; j < 4; ++j)
        wcol[j] = W + (size_t)(2 * half) * N + n0 + 16 * j + lo;

    v8f acc[4][4] = {};

    for (int k = 0; k < K; k += 4) {
        if ((k & 31) == 0) {
            // stream-ahead prefetch of the A rows (global_prefetch_b8)
#pragma unroll
            for (int i = 0; i < 4; ++i)
                __builtin_prefetch(arow[i] + k + 32, 0, 0);
        }
        v2f a[4], b[4];
#pragma unroll
        for (int i = 0; i < 4; ++i)
            a[i] = *(const v2f*)(arow[i] + k);
#pragma unroll
        for (int j = 0; j < 4; ++j) {
            b[j].x = wcol[j][(size_t)k * N];
            b[j].y = wcol[j][(size_t)k * N + N];
        }
#pragma unroll
        for (int i = 0; i < 4; ++i)
#pragma unroll
            for (int j = 0; j < 4; ++j)
                acc[i][j] = __builtin_amdgcn_wmma_f32_16x16x4_f32(
                    false, a[i], false, b[j], (short)0, acc[i][j], false, false);
    }

    float bv[4];
#pragma unroll
    for (int j = 0; j < 4; ++j) bv[j] = bias[n0 + 16 * j + lo];

#pragma unroll
    for (int i = 0; i < 4; ++i) {
        float* crow = C + (size_t)(m0 + 16 * i + 8 * half) * N + n0 + lo;
#pragma unroll
        for (int r = 0; r < 8; ++r)
#pragma unroll
            for (int j = 0; j < 4; ++j)
                crow[(size_t)r * N + 16 * j] = acc[i][j][r] + bv[j];
    }
}

// ---------------------------------------------------------------------------
// RMS-norm (over the full 1024-wide q / k rows) + rotary, in place on qkv.
// One 256-thread block per token row.
// ---------------------------------------------------------------------------
__global__ __launch_bounds__(256)
void norm_rope(float* __restrict__ qkv, const float* __restrict__ rot,
               const float* __restrict__ wq, const float* __restrict__ wk) {
    const int row = blockIdx.x;          // b*SEQ + s
    float* q = qkv + (size_t)row * QKV_N;
    float* k = q + INNER;
    const float* fr = rot + (size_t)row * (2 * HEAD_DIM);
    const int t = threadIdx.x;

    float sq = 0.f, sk = 0.f;
    for (int i = t; i < INNER; i += 256) {
        float a = q[i]; sq += a * a;
        float b = k[i]; sk += b * b;
    }
    __shared__ float redq[256], redk[256];
    redq[t] = sq; redk[t] = sk;
    __syncthreads();
    for (int off = 128; off > 0; off >>= 1) {
        if (t < off) { redq[t] += redq[t + off]; redk[t] += redk[t + off]; }
        __syncthreads();
    }
    const float invq = rsqrtf(redq[0] * (1.0f / INNER) + 1e-5f);
    const float invk = rsqrtf(redk[0] * (1.0f / INNER) + 1e-5f);

    // 512 rotation pairs per section; freqs: cos_e[i]=fr[2i], sin_o[i]=fr[64+2i+1]
    for (int p = t; p < INNER / 2; p += 256) {
        const int i2  = p & 31;                 // pair index within head
        const float c = fr[2 * i2];
        const float s = fr[HEAD_DIM + 2 * i2 + 1];
        const int idx = 2 * p;

        float xe = q[idx]     * invq * wq[idx];
        float xo = q[idx + 1] * invq * wq[idx + 1];
        q[idx]     = xe * c - xo * s;
        q[idx + 1] = xe * s + xo * c;

        xe = k[idx]     * invk * wk[idx];
        xo = k[idx + 1] * invk * wk[idx + 1];
        k[idx]     = xe * c - xo * s;
        k[idx + 1] = xe * s + xo * c;
    }
}

// ---------------------------------------------------------------------------
// Flash attention: one wave per (b, h, 16-query tile). Online softmax in
// registers, P re-layout C->A via 1KB LDS bounce, all GEMMs via fp32 WMMA.
// History-key scaling folded into the score scale.
// ---------------------------------------------------------------------------
__global__ __launch_bounds__(32)
void attn_kernel(const float* __restrict__ qkv, const float* __restrict__ hks,
                 const int* __restrict__ oclp, float* __restrict__ out) {
    __shared__ __align__(16) float plds[256];

    const int bx = blockIdx.x;
    const int qt = bx & (QTILES - 1);
    const int h  = (bx >> 7) & (NUM_HEADS - 1);
    const int b  = bx >> 11;

    const int l    = threadIdx.x;
    const int lo   = l & 15;
    const int half = l >> 4;

    const int   hist = SEQ - *oclp;
    const float skh  = 1.0f + 9.0f / (1.0f + __expf(-hks[h]));
    const float sscale = 0.125f;   // 1/sqrt(64)

    const size_t base  = (size_t)b * SEQ * QKV_N;
    const float* qbase = qkv + base + (size_t)(qt * 16 + lo) * QKV_N + h * HEAD_DIM + 2 * half;
    const float* kbase = qkv + base + INNER     + h * HEAD_DIM + 2 * half;
    const float* vbase = qkv + base + 2 * INNER + h * HEAD_DIM;

    // Q fragments: qa[d] covers K dims [4d, 4d+4)
    v2f qa[16];
#pragma unroll
    for (int d = 0; d < 16; ++d)
        qa[d] = *(const v2f*)(qbase + 4 * d);

    v8f o[4] = {};
    float m_state[8], l_state[8];
#pragma unroll
    for (int r = 0; r < 8; ++r) { m_state[r] = -__builtin_inff(); l_state[r] = 0.f; }

    for (int t = 0; t < QTILES; ++t) {
        // --- scores: S = Q @ K^T (16 chained fp32 WMMAs over D=64) ---
        v8f c = {};
        const float* krow = kbase + (size_t)(t * 16 + lo) * QKV_N;
#pragma unroll
        for (int d = 0; d < 16; ++d) {
            v2f kb = *(const v2f*)(krow + 4 * d);
            c = __builtin_amdgcn_wmma_f32_16x16x4_f32(false, qa[d], false, kb,
                                                      (short)0, c, false, false);
        }

        const int   kg  = t * 16 + lo;          // this lane's key column (global)
        const float fac = sscale * (kg < hist ? skh : 1.0f);

        float p[8], alpha[8];
#pragma unroll
        for (int r = 0; r < 8; ++r) {
            float s = c[r] * fac;
            float rm = s;                        // row max (16-lane butterfly)
            rm = fmaxf(rm, __shfl_xor(rm, 1));
            rm = fmaxf(rm, __shfl_xor(rm, 2));
            rm = fmaxf(rm, __shfl_xor(rm, 4));
            rm = fmaxf(rm, __shfl_xor(rm, 8));
            const float mn = fmaxf(m_state[r], rm);
            const float al = __expf(m_state[r] - mn);
            const float pv = __expf(s - mn);
            float su = pv;                       // row sum
            su += __shfl_xor(su, 1);
            su += __shfl_xor(su, 2);
            su += __shfl_xor(su, 4);
            su += __shfl_xor(su, 8);
            l_state[r] = l_state[r] * al + su;
            m_state[r] = mn;
            alpha[r] = al;
            p[r] = pv;
        }

        // rescale running output by alpha (per row r+8*half)
#pragma unroll
        for (int nt = 0; nt < 4; ++nt)
#pragma unroll
            for (int r = 0; r < 8; ++r)
                o[nt][r] *= alpha[r];

        // P: C-layout -> A-layout through LDS
#pragma unroll
        for (int r = 0; r < 8; ++r)
            plds[(r + 8 * half) * 16 + lo] = p[r];
        __syncthreads();
        v2f pa[4];
#pragma unroll
        for (int kk = 0; kk < 4; ++kk) {
            pa[kk].x = plds[lo * 16 + 4 * kk + 2 * half];
            pa[kk].y = plds[lo * 16 + 4 * kk + 2 * half + 1];
        }
        __syncthreads();

        // --- O += P @ V (4 K-steps x 4 N-tiles of fp32 WMMA) ---
        const float* vt = vbase + (size_t)(t * 16) * QKV_N;
#pragma unroll
        for (int kk = 0; kk < 4; ++kk) {
            const int kr = 4 * kk + 2 * half;
#pragma unroll
            for (int nt = 0; nt < 4; ++nt) {
                v2f vb;
                vb.x = vt[(size_t)kr * QKV_N + nt * 16 + lo];
                vb.y = vt[(size_t)(kr + 1) * QKV_N + nt * 16 + lo];
                o[nt] = __builtin_amdgcn_wmma_f32_16x16x4_f32(false, pa[kk], false, vb,
                                                              (short)0, o[nt], false, false);
            }
        }
    }

    // normalize and store: out[b, q, h, d] with row stride INNER
    float inv[8];
#pragma unroll
    for (int r = 0; r < 8; ++r) inv[r] = 1.0f / l_state[r];

    float* orow = out + (size_t)((size_t)b * SEQ + qt * 16 + 8 * half) * INNER
                      + h * HEAD_DIM + lo;
#pragma unroll
    for (int nt = 0; nt < 4; ++nt)
#pragma unroll
        for (int r = 0; r < 8; ++r)
            orow[(size_t)r * INNER + nt * 16] = o[nt][r] * inv[r];
}

// ---------------------------------------------------------------------------
extern "C" void kernel_launch(void* const* d_in, const int* in_sizes, int n_in,
                              void* d_out, int out_size, void* d_ws, size_t ws_size,
                              hipStream_t stream) {
    const float* hidden = (const float*)d_in[0];
    const float* rot    = (const float*)d_in[1];
    const float* w_qkv  = (const float*)d_in[2];
    const float* b_qkv  = (const float*)d_in[3];
    const float* nqw    = (const float*)d_in[4];
    const float* nkw    = (const float*)d_in[5];
    const float* w_out  = (const float*)d_in[6];
    const float* b_out  = (const float*)d_in[7];
    const float* hks    = (const float*)d_in[8];
    const int*   ocl    = (const int*)d_in[9];
    float* out = (float*)d_out;

    float* qkv  = (float*)d_ws;                       // [4096, 3072]  50.3 MB
    float* attn = qkv + (size_t)BS * QKV_N;           // [4096, 1024]  16.8 MB

    // 1) qkv = hidden @ w_qkv + b_qkv   (64x64 tile per wave)
    gemm_bias_f32<<<dim3(QKV_N / 64, BS / 64), 32, 0, stream>>>(
        hidden, w_qkv, b_qkv, qkv, BS, QKV_N, DIM);

    // 2) RMS-norm + RoPE in place on q,k sections
    norm_rope<<<BS, 256, 0, stream>>>(qkv, rot, nqw, nkw);

    // 3) flash attention
    attn_kernel<<<BATCH * NUM_HEADS * QTILES, 32, 0, stream>>>(qkv, hks, ocl, attn);

    // 4) d_out = attn @ w_out + b_out  (64x64 tile per wave)
    gemm_bias_f32<<<dim3(DIM / 64, BS / 64), 32, 0, stream>>>(
        attn, w_out, b_out, out, BS, DIM, DIM);
}